// LocalCrossCorrelation2D_Uncertainty_87582973100654
// MI455X (gfx1250) — compile-verified
//
#include <hip/hip_runtime.h>
#include <hip/hip_bf16.h>

typedef __attribute__((ext_vector_type(2))) float v2f;
typedef __attribute__((ext_vector_type(8))) float v8f;

#define H_IMG 512
#define W_IMG 512
#define NPIX (H_IMG * W_IMG)               // 262144
#define TILES_X (W_IMG / 16)               // 32
#define TILES_PER_IMG (TILES_X * (H_IMG/16)) // 1024
#define NBATCH 32
#define NTILES (NBATCH * TILES_PER_IMG)    // 32768
#define WAVES_PER_BLOCK 8
#define LDS_ROWS 24
#define LDS_STRIDE 40                      // 24x32 staged, pad to 40 dwords -> conflict-free halves

// Stage-1: per-wave 16x16 tile. Box sums via horizontal VALU pass (B-fragment
// layout) + vertical banded matmul on V_WMMA_F32_16X16X4_F32 (K=24 as 6 chunks).
__global__ __launch_bounds__(256) void lcc_tile_kernel(
    const float* __restrict__ I, const float* __restrict__ J,
    const float* __restrict__ U, float* __restrict__ out,
    float* __restrict__ part)
{
    __shared__ float sI[WAVES_PER_BLOCK][LDS_ROWS * LDS_STRIDE];
    __shared__ float sJ[WAVES_PER_BLOCK][LDS_ROWS * LDS_STRIDE];

    const int lane = threadIdx.x & 31;
    const int wave = threadIdx.x >> 5;
    const int t    = blockIdx.x * WAVES_PER_BLOCK + wave;   // global tile id
    const int b    = t >> 10;                               // batch
    const int ti   = t & (TILES_PER_IMG - 1);
    const int R    = (ti >> 5) << 4;                        // tile row origin
    const int C    = (ti & 31) << 4;                        // tile col origin

    const float* __restrict__ Ib = I + (size_t)b * NPIX;
    const float* __restrict__ Jb = J + (size_t)b * NPIX;
    const float* __restrict__ Ub = U + (size_t)b * NPIX;

    float* tI = &sI[wave][0];
    float* tJ = &sJ[wave][0];

    // ---- Stage 24x32 halo tile (zero padding outside image) ----
    {
        const int gc  = C - 4 + lane;                       // cols C-4 .. C+27
        const bool cok = (gc >= 0) & (gc < W_IMG);
        #pragma unroll
        for (int r = 0; r < LDS_ROWS; ++r) {
            const int gr = R - 4 + r;
            float vi = 0.0f, vj = 0.0f;
            if (cok & (gr >= 0) & (gr < H_IMG)) {
                const size_t idx = (size_t)gr * W_IMG + gc;
                vi = Ib[idx];
                vj = Jb[idx];
            }
            tI[r * LDS_STRIDE + lane] = vi;
            tJ[r * LDS_STRIDE + lane] = vj;
        }
    }
    asm volatile("s_wait_dscnt 0" ::: "memory");  // wave-private LDS: order stage->consume

    // ---- Box sums: horizontal 9-sum in VALU, vertical 9-sum via WMMA ----
    v8f accI = {}, accJ = {}, accII = {}, accJJ = {}, accIJ = {};
    const int m   = lane & 15;            // A-matrix row M / B-matrix col N
    const int off = (lane >> 4) << 1;     // lanes 0-15 -> K+{0,1}; lanes 16-31 -> K+{2,3}

    #pragma unroll
    for (int kc = 0; kc < 6; ++kc) {
        const int kb = kc * 4;
        v2f bI, bJ, bII, bJJ, bIJ, aF;
        #pragma unroll
        for (int p = 0; p < 2; ++p) {
            const int r = kb + off + p;   // staged input row == H row (K index)
            float hI = 0.f, hJ = 0.f, hII = 0.f, hJJ = 0.f, hIJ = 0.f;
            const int base = r * LDS_STRIDE + m;   // H column n = lane&15
            #pragma unroll
            for (int q = 0; q < 9; ++q) {
                const float x = tI[base + q];
                const float y = tJ[base + q];
                hI += x; hJ += y;
                hII = __builtin_fmaf(x, x, hII);
                hJJ = __builtin_fmaf(y, y, hJJ);
                hIJ = __builtin_fmaf(x, y, hIJ);
            }
            bI[p] = hI; bJ[p] = hJ; bII[p] = hII; bJJ[p] = hJJ; bIJ[p] = hIJ;
            const int d = r - m;          // band matrix T[m][k] = (0 <= k-m <= 8)
            aF[p] = (d >= 0 && d <= 8) ? 1.0f : 0.0f;
        }
        accI  = __builtin_amdgcn_wmma_f32_16x16x4_f32(false, aF, false, bI,  (short)0, accI,  false, false);
        accJ  = __builtin_amdgcn_wmma_f32_16x16x4_f32(false, aF, false, bJ,  (short)0, accJ,  false, false);
        accII = __builtin_amdgcn_wmma_f32_16x16x4_f32(false, aF, false, bII, (short)0, accII, false, false);
        accJJ = __builtin_amdgcn_wmma_f32_16x16x4_f32(false, aF, false, bJJ, (short)0, accJJ, false, false);
        accIJ = __builtin_amdgcn_wmma_f32_16x16x4_f32(false, aF, false, bIJ, (short)0, accIJ, false, false);
    }

    // ---- NCC epilogue: D layout -> lane holds (M = v + 8*(lane>=16), N = lane&15) ----
    const float wsz    = 81.0f;
    const float inv_ws = 1.0f / 81.0f;
    const float EPSV   = 3.0590232050182579e-07f;   // e^-15

    float s1 = 0.0f, s2 = 0.0f;
    float* __restrict__ out_cc  = out + 64;
    float* __restrict__ out_ccu = out + 64 + (size_t)NBATCH * NPIX;
    const size_t obase = (size_t)b * NPIX;

    #pragma unroll
    for (int v = 0; v < 8; ++v) {
        const int mr = v + ((lane >> 4) << 3);
        const int n  = lane & 15;
        const size_t idx = (size_t)(R + mr) * W_IMG + (C + n);

        const float Is  = accI[v],  Js  = accJ[v];
        const float IIs = accII[v], JJs = accJJ[v], IJs = accIJ[v];

        const float uI = Is * inv_ws;
        const float uJ = Js * inv_ws;
        float cross = IJs - uJ * Is - uI * Js + uI * uJ * wsz;
        float Iv    = IIs - 2.0f * uI * Is + uI * uI * wsz;
        float Jv    = JJs - 2.0f * uJ * Js + uJ * uJ * wsz;

        if (!((Iv * Jv) > EPSV)) { cross = 1.0f; Iv = 1.0f; Jv = 1.0f; }
        const float cc = (cross * cross) / (Iv * Jv + EPSV);

        const float u = Ub[idx];
        out_cc[obase + idx]  = cc;
        out_ccu[obase + idx] = cc * u;
        s1 += (1.0f - cc);
        s2 += (1.0f - cc) * u;
    }

    // wave32 butterfly reduction
    #pragma unroll
    for (int d = 16; d >= 1; d >>= 1) {
        s1 += __shfl_xor(s1, d, 32);
        s2 += __shfl_xor(s2, d, 32);
    }
    if (lane == 0) {
        part[2 * t]     = s1;
        part[2 * t + 1] = s2;
    }
}

// Stage-2: deterministic per-batch reduction of 1024 tile partials.
__global__ __launch_bounds__(256) void lcc_reduce_kernel(
    const float* __restrict__ part, float* __restrict__ out)
{
    __shared__ float r1[256];
    __shared__ float r2[256];
    const int b   = blockIdx.x;
    const int tid = threadIdx.x;

    float s1 = 0.0f, s2 = 0.0f;
    for (int i = tid; i < TILES_PER_IMG; i += 256) {
        const int k = 2 * (b * TILES_PER_IMG + i);
        s1 += part[k];
        s2 += part[k + 1];
    }
    r1[tid] = s1; r2[tid] = s2;
    __syncthreads();
    #pragma unroll
    for (int s = 128; s > 0; s >>= 1) {
        if (tid < s) { r1[tid] += r1[tid + s]; r2[tid] += r2[tid + s]; }
        __syncthreads();
    }
    if (tid == 0) {
        const float inv_n = 1.0f / (float)NPIX;
        out[b]          = r1[0] * inv_n;   // loss
        out[NBATCH + b] = r2[0] * inv_n;   // loss_u
    }
}

extern "C" void kernel_launch(void* const* d_in, const int* in_sizes, int n_in,
                              void* d_out, int out_size, void* d_ws, size_t ws_size,
                              hipStream_t stream) {
    const float* I = (const float*)d_in[0];
    const float* J = (const float*)d_in[1];
    const float* U = (const float*)d_in[2];
    float* out  = (float*)d_out;
    float* part = (float*)d_ws;   // needs 32768*2*4 = 256 KB

    lcc_tile_kernel<<<NTILES / WAVES_PER_BLOCK, 256, 0, stream>>>(I, J, U, out, part);
    lcc_reduce_kernel<<<NBATCH, 256, 0, stream>>>(part, out);
}